// CompGraphConv_84817014161721
// MI455X (gfx1250) — compile-verified
//
#include <hip/hip_runtime.h>

typedef __attribute__((ext_vector_type(2))) float v2f;
typedef __attribute__((ext_vector_type(8))) float v8f;

#define N_NODES 100000
#define N_EDGES 1000000
#define HID 128

// LDS A-tile geometry: 16 rows x 128 cols, row stride padded to 132 floats
// (132 mod 64 banks = 4 -> lanes 0-15 hit banks 0,4,...,60: conflict-free b64)
#define TROWS 16
#define RSTRIDE 132
#define MATSZ (TROWS * RSTRIDE)   // 2112 floats per matrix tile

// ---------------------------------------------------------------------------
// 0) zero workspace (agg_f, agg_r, deg_f, deg_r)
// ---------------------------------------------------------------------------
__global__ void zero_ws_kernel(float* __restrict__ ws, long n) {
    long i = (long)blockIdx.x * blockDim.x + threadIdx.x;
    long stride = (long)gridDim.x * blockDim.x;
    for (; i < n; i += stride) ws[i] = 0.0f;
}

// ---------------------------------------------------------------------------
// 1) edge scatter: one wave32 per edge.
//    agg_f[dst] += n_feat[src];  deg_f[dst] += 1
//    agg_r[src] += n_feat[dst];  deg_r[src] += 1
//    (the "- e_vec" term is folded into the normalize pass via deg*e_vec)
// ---------------------------------------------------------------------------
__global__ __launch_bounds__(256)
void scatter_kernel(const float* __restrict__ n_feat,
                    const int* __restrict__ src,
                    const int* __restrict__ dst,
                    float* __restrict__ agg_f,
                    float* __restrict__ agg_r,
                    float* __restrict__ deg_f,
                    float* __restrict__ deg_r) {
    long gtid = (long)blockIdx.x * blockDim.x + threadIdx.x;
    long edge = gtid >> 5;              // one wave32 per edge
    int  lane = threadIdx.x & 31;       // lane handles 4 channels
    if (edge >= N_EDGES) return;

    int s = src[edge];
    int d = dst[edge];

    const float4* fs = (const float4*)(n_feat + (long)s * HID);
    const float4* fd = (const float4*)(n_feat + (long)d * HID);
    float4 vs = fs[lane];               // n_feat[s], channels 4*lane..4*lane+3
    float4 vd = fd[lane];               // n_feat[d]

    float* of  = agg_f + (long)d * HID + lane * 4;
    float* orv = agg_r + (long)s * HID + lane * 4;
    atomicAdd(of + 0, vs.x);
    atomicAdd(of + 1, vs.y);
    atomicAdd(of + 2, vs.z);
    atomicAdd(of + 3, vs.w);
    atomicAdd(orv + 0, vd.x);
    atomicAdd(orv + 1, vd.y);
    atomicAdd(orv + 2, vd.z);
    atomicAdd(orv + 3, vd.w);
    if (lane == 0) {
        atomicAdd(deg_f + d, 1.0f);
        atomicAdd(deg_r + s, 1.0f);
    }
}

// ---------------------------------------------------------------------------
// 2) normalize: comp = (agg - deg*e_vec) * max(deg,1)^-0.5 * (2/3)   (in place)
// ---------------------------------------------------------------------------
__global__ void normalize_kernel(const float* __restrict__ e_fwd,
                                 const float* __restrict__ e_rev,
                                 float* __restrict__ agg_f,
                                 float* __restrict__ agg_r,
                                 const float* __restrict__ deg_f,
                                 const float* __restrict__ deg_r) {
    int n = blockIdx.x * blockDim.x + threadIdx.x;
    if (n >= N_NODES) return;
    float df = deg_f[n];
    float dr = deg_r[n];
    const float k23 = 2.0f / 3.0f;
    float nf = k23 / sqrtf(fmaxf(df, 1.0f));
    float nr = k23 / sqrtf(fmaxf(dr, 1.0f));

    float4* af = (float4*)(agg_f + (long)n * HID);
    float4* ar = (float4*)(agg_r + (long)n * HID);
    const float4* ef = (const float4*)e_fwd;
    const float4* er = (const float4*)e_rev;
#pragma unroll 4
    for (int i = 0; i < HID / 4; ++i) {
        float4 a = af[i], e = ef[i];
        a.x = (a.x - df * e.x) * nf;
        a.y = (a.y - df * e.y) * nf;
        a.z = (a.z - df * e.z) * nf;
        a.w = (a.w - df * e.w) * nf;
        af[i] = a;
        float4 b = ar[i], g = er[i];
        b.x = (b.x - dr * g.x) * nr;
        b.y = (b.y - dr * g.y) * nr;
        b.z = (b.z - dr * g.z) * nr;
        b.w = (b.w - dr * g.w) * nr;
        ar[i] = b;
    }
}

// ---------------------------------------------------------------------------
// 3) fused node GEMM via V_WMMA_F32_16X16X4_F32, A tiles staged in LDS:
//    out = comp_f@W_I^T + comp_r@W_O^T + (1/3)*n_feat@W_S^T + bias_comb
//    (comp_f/comp_r already carry the 2/3 factor; 1/3 folded into LDS stage)
//    Block = 8 wave32s; block owns rows [16b,16b+16), wave w owns cols
//    [16w,16w+16). 96 WMMAs per wave; A fragments via ds_load_b64.
// ---------------------------------------------------------------------------
__global__ __launch_bounds__(256)
void gemm_kernel(const float* __restrict__ comp_f,
                 const float* __restrict__ comp_r,
                 const float* __restrict__ n_feat,
                 const float* __restrict__ W_I,
                 const float* __restrict__ W_O,
                 const float* __restrict__ W_S,
                 const float* __restrict__ b_I,
                 const float* __restrict__ b_O,
                 const float* __restrict__ b_S,
                 float* __restrict__ out) {
    __shared__ float ldsA[3 * MATSZ];   // ~25 KB of the 320 KB WGP LDS

    const int tid    = threadIdx.x;
    const int wave   = tid >> 5;        // 0..7 -> N tile
    const int lane   = tid & 31;
    const int half   = lane >> 4;       // 0: K lo pair, 1: K hi pair
    const int lid    = lane & 15;
    const int m_base = blockIdx.x * 16; // 6250 * 16 == 100000 exactly
    const int n_base = wave * 16;

    const float* As[3] = {comp_f, comp_r, n_feat};
    const float* Ws[3] = {W_I, W_O, W_S};

    // --- stage the 16x128 A tile of each matrix into LDS (coalesced b128) ---
    // 512 float4 per matrix tile; 256 threads -> 2 float4 each
#pragma unroll
    for (int mat = 0; mat < 3; ++mat) {
        const float4* Ag = (const float4*)(As[mat] + (long)m_base * HID);
        const float scl = (mat == 2) ? (1.0f / 3.0f) : 1.0f;
#pragma unroll
        for (int i = 0; i < 2; ++i) {
            int idx  = tid + i * 256;         // float4 index in 16x128 tile
            int row  = idx >> 5;              // 32 float4 per row
            int col4 = idx & 31;
            float4 v = Ag[idx];
            v.x *= scl; v.y *= scl; v.z *= scl; v.w *= scl;
            *(float4*)(ldsA + mat * MATSZ + row * RSTRIDE + col4 * 4) = v;
        }
    }
    __syncthreads();

    v8f acc = {};
#pragma unroll
    for (int mat = 0; mat < 3; ++mat) {
        const float* Al = ldsA + mat * MATSZ + lid * RSTRIDE; // A row m=lid
        const float* W  = Ws[mat] + (long)(n_base + lid) * HID; // B col n=lid
#pragma unroll 8
        for (int kk = 0; kk < HID; kk += 4) {
            const int ko = kk + 2 * half;
            // A 16x4 f32 layout: lanes 0-15 K={kk,kk+1}, lanes 16-31 K={kk+2,kk+3}
            v2f a = *(const v2f*)(Al + ko);   // ds_load_b64
            // B 4x16: B[k][n] = W[n][k]; mirrors A layout per half-wave
            v2f b;
            b.x = W[ko];
            b.y = W[ko + 1];
#if __has_builtin(__builtin_amdgcn_wmma_f32_16x16x4_f32)
            acc = __builtin_amdgcn_wmma_f32_16x16x4_f32(
                false, a, false, b, (short)0, acc, false, false);
#else
            // scalar fallback (should not be taken on gfx1250 toolchains)
            for (int r = 0; r < 8; ++r) acc[r] += a.x * b.x + a.y * b.y;
#endif
        }
    }

    const int col = n_base + lid;
    const float bias = (b_I[col] + b_O[col]) * (2.0f / 3.0f)
                     + b_S[col] * (1.0f / 3.0f);
    // C/D 16x16 f32 layout: lanes 0-15 M=r, lanes 16-31 M=8+r, N=lid
#pragma unroll
    for (int r = 0; r < 8; ++r) {
        const int row = m_base + half * 8 + r;
        out[(long)row * HID + col] = acc[r] + bias;
    }
}

// ---------------------------------------------------------------------------
// 4) edge-feature update: e_out = e_vec @ W_rel^T + b_rel  (two 128-GEMVs)
// ---------------------------------------------------------------------------
__global__ void erel_kernel(const float* __restrict__ e_fwd,
                            const float* __restrict__ e_rev,
                            const float* __restrict__ W_rel,
                            const float* __restrict__ b_rel,
                            float* __restrict__ out) {
    int j = blockIdx.x * blockDim.x + threadIdx.x;
    if (j >= HID) return;
    const float* w = W_rel + (long)j * HID;
    float sf = 0.0f, sr = 0.0f;
#pragma unroll 8
    for (int k = 0; k < HID; ++k) {
        sf = fmaf(e_fwd[k], w[k], sf);
        sr = fmaf(e_rev[k], w[k], sr);
    }
    out[j] = sf + b_rel[j];
    out[HID + j] = sr + b_rel[j];
}

// ---------------------------------------------------------------------------
extern "C" void kernel_launch(void* const* d_in, const int* in_sizes, int n_in,
                              void* d_out, int out_size, void* d_ws, size_t ws_size,
                              hipStream_t stream) {
    const float* n_feat = (const float*)d_in[0];
    const float* e_fwd  = (const float*)d_in[1];
    const float* e_rev  = (const float*)d_in[2];
    const float* W_I    = (const float*)d_in[3];
    const float* b_I    = (const float*)d_in[4];
    const float* W_O    = (const float*)d_in[5];
    const float* b_O    = (const float*)d_in[6];
    const float* W_S    = (const float*)d_in[7];
    const float* b_S    = (const float*)d_in[8];
    const float* W_rel  = (const float*)d_in[9];
    const float* b_rel  = (const float*)d_in[10];
    const int*   src    = (const int*)d_in[11];
    const int*   dst    = (const int*)d_in[12];

    float* out = (float*)d_out;
    float* ws  = (float*)d_ws;

    float* agg_f = ws;                                  // N*128
    float* agg_r = ws + (long)N_NODES * HID;            // N*128
    float* deg_f = ws + 2L * N_NODES * HID;             // N
    float* deg_r = deg_f + N_NODES;                     // N

    const long zero_n = 2L * N_NODES * HID + 2L * N_NODES;
    zero_ws_kernel<<<2048, 256, 0, stream>>>(ws, zero_n);

    // 1M edges, one wave32 per edge, 8 waves / block
    scatter_kernel<<<N_EDGES / 8, 256, 0, stream>>>(
        n_feat, src, dst, agg_f, agg_r, deg_f, deg_r);

    normalize_kernel<<<(N_NODES + 255) / 256, 256, 0, stream>>>(
        e_fwd, e_rev, agg_f, agg_r, deg_f, deg_r);

    // 100000 / 16 = 6250 blocks, each 8 wave32s covering 16x128 output
    gemm_kernel<<<N_NODES / 16, 256, 0, stream>>>(
        agg_f, agg_r, n_feat, W_I, W_O, W_S, b_I, b_O, b_S, out);

    erel_kernel<<<1, HID, 0, stream>>>(
        e_fwd, e_rev, W_rel, b_rel, out + (long)N_NODES * HID);
}